// BilinearInterpolation_91225105367442
// MI455X (gfx1250) — compile-verified
//
#include <hip/hip_runtime.h>
#include <hip/hip_bf16.h>

typedef __attribute__((ext_vector_type(2))) float v2f;
typedef __attribute__((ext_vector_type(8))) float v8f;

#define BATCH 16
#define CH    64
#define HIN   256
#define WIN   256
#define HOUT  256
#define WOUT  256

// out[b][c][j][i] = bilinear(X[b][c], x(i,j), y(i,j))
// x = 128*(t00*xg + t01*yg + t02) + 128,  xg = -1 + i*2/255, yg = -1 + j*2/255
__global__ __launch_bounds__(256) void BilinearInterpolation_91225105367442_kernel(
    const float* __restrict__ X,
    const float* __restrict__ theta,   // [B,6]
    float* __restrict__ out)
{
    __shared__ float xs[WOUT];
    __shared__ float ys[WOUT];

    const int tid  = threadIdx.x;
    const int lane = tid & 31;
    const int wave = tid >> 5;
    const int j = blockIdx.x;   // output row (y index)
    const int b = blockIdx.y;   // batch

    // ---------- Phase 1: coordinates via V_WMMA_F32_16X16X4_F32 ----------
    // D(16x16) = A(16x4) * B(4x16) + 0
    //   A row m  = [xg(T+m), yg(j), 1, 0]
    //   B col n  : batch bb=(b&~7)+(n>>1), coord c=n&1 ->
    //              [128*t_c0, 128*t_c1, 128*t_c2+128, 0]
    //   D[m][n]  = image-space coordinate c of batch bb at column i=T+m
    const int n    = lane & 15;   // column of B / D
    const int hi   = lane >> 4;   // 0 -> K={0,1}, 1 -> K={2,3} (A & B striping)
    const int bb   = (b & ~7) + (n >> 1);
    const int cidx = n & 1;
    const float t0 = theta[bb * 6 + cidx * 3 + 0];
    const float t1 = theta[bb * 6 + cidx * 3 + 1];
    const float t2 = theta[bb * 6 + cidx * 3 + 2];
    v2f Bm;
    Bm[0] = hi ? fmaf(t2, 128.0f, 128.0f) : (t0 * 128.0f);  // K=0 | K=2
    Bm[1] = hi ? 0.0f                     : (t1 * 128.0f);  // K=1 | K=3
    const float yg = fmaf((float)j, 2.0f / 255.0f, -1.0f);

    #pragma unroll
    for (int t = 0; t < 2; ++t) {
        const int T = wave * 32 + t * 16;          // tile of 16 output columns
        const int m = lane & 15;                   // A row
        const float xg = fmaf((float)(T + m), 2.0f / 255.0f, -1.0f);
        v2f Am;
        Am[0] = hi ? 1.0f : xg;                    // K=0 | K=2
        Am[1] = hi ? 0.0f : yg;                    // K=1 | K=3
        v8f Cm = {};
        v8f D = __builtin_amdgcn_wmma_f32_16x16x4_f32(
            /*neg_a=*/false, Am, /*neg_b=*/false, Bm,
            /*c_mod=*/(short)0, Cm, /*reuse_a=*/false, /*reuse_b=*/false);
        // D layout: VGPR r, lane l -> M = r + 8*(l>>4), N = l&15.
        // Scatter the two columns belonging to this block's batch to LDS.
        const int want = (b & 7) * 2;
        if (n == want || n == want + 1) {
            float* dst = (n == want) ? xs : ys;
            const int mb = T + hi * 8;
            #pragma unroll
            for (int r = 0; r < 8; ++r) dst[mb + r] = D[r];
        }
    }
    __syncthreads();

    // ---------- Phase 2: bilinear gather (memory bound) ----------
    const int i   = tid;
    const float x = xs[i];
    const float y = ys[i];

    int x0 = (int)floorf(x);
    int y0 = (int)floorf(y);
    int x1 = x0 + 1;
    int y1 = y0 + 1;
    x0 = min(max(x0, 0), WIN - 1);
    x1 = min(max(x1, 0), WIN - 1);
    y0 = min(max(y0, 0), HIN - 1);
    y1 = min(max(y1, 0), HIN - 1);

    const float x0f = (float)x0, x1f = (float)x1;
    const float y0f = (float)y0, y1f = (float)y1;
    const float wa = (x1f - x) * (y1f - y);
    const float wb = (x1f - x) * (y - y0f);
    const float wc = (x - x0f) * (y1f - y);
    const float wd = (x - x0f) * (y - y0f);

    const size_t HW = (size_t)HIN * WIN;
    const float* Xb = X + (size_t)b * CH * HW;
    const int iA = y0 * WIN + x0;
    const int iB = y1 * WIN + x0;
    const int iC = y0 * WIN + x1;
    const int iD = y1 * WIN + x1;

    float* op = out + (size_t)b * CH * (size_t)HOUT * WOUT
                    + (size_t)j * WOUT + i;

    #pragma unroll 8
    for (int c = 0; c < CH; ++c) {
        const float* p = Xb + (size_t)c * HW;
        float v = wa * p[iA] + wb * p[iB] + wc * p[iC] + wd * p[iD];
        // streamed output: non-temporal so L2 stays dedicated to the X gather
        __builtin_nontemporal_store(v, op + (size_t)c * HOUT * WOUT);
    }
}

extern "C" void kernel_launch(void* const* d_in, const int* in_sizes, int n_in,
                              void* d_out, int out_size, void* d_ws, size_t ws_size,
                              hipStream_t stream) {
    const float* X     = (const float*)d_in[0];   // [16,64,256,256] f32
    const float* theta = (const float*)d_in[1];   // [16,6] f32
    float* out = (float*)d_out;                   // [16,64,256,256] f32
    (void)in_sizes; (void)n_in; (void)out_size; (void)d_ws; (void)ws_size;

    dim3 grid(HOUT, BATCH);   // one block per (output row j, batch b)
    dim3 block(256);          // 8 wave32 per block, thread i = output column
    BilinearInterpolation_91225105367442_kernel<<<grid, block, 0, stream>>>(X, theta, out);
}